// HybridQCNN_953482739911
// MI455X (gfx1250) — compile-verified
//
#include <hip/hip_runtime.h>
#include <math.h>

typedef __attribute__((ext_vector_type(2))) float v2f;
typedef __attribute__((ext_vector_type(8))) float v8f;

#define LSTRIDE 33

// ================= Precompute: build U (32x32) then G_c (2x32x32) =================
__device__ inline void apply1(float* re, float* im, int w,
                              float g00r, float g00i, float g01r, float g01i,
                              float g10r, float g10i, float g11r, float g11i) {
  int mask = 1 << (4 - w);
  for (int i = 0; i < 32; ++i) {
    if (i & mask) continue;
    int j = i | mask;
    float ar = re[i], ai = im[i], br = re[j], bi = im[j];
    re[i] = g00r*ar - g00i*ai + g01r*br - g01i*bi;
    im[i] = g00r*ai + g00i*ar + g01r*bi + g01i*br;
    re[j] = g10r*ar - g10i*ai + g11r*br - g11i*bi;
    im[j] = g10r*ai + g10i*ar + g11r*bi + g11i*br;
  }
}
__device__ inline void applyc1(float* re, float* im, int cw, int tw,
                               float g00r, float g00i, float g01r, float g01i,
                               float g10r, float g10i, float g11r, float g11i) {
  int cm = 1 << (4 - cw), tm = 1 << (4 - tw);
  for (int i = 0; i < 32; ++i) {
    if (!(i & cm) || (i & tm)) continue;
    int j = i | tm;
    float ar = re[i], ai = im[i], br = re[j], bi = im[j];
    re[i] = g00r*ar - g00i*ai + g01r*br - g01i*bi;
    im[i] = g00r*ai + g00i*ar + g01r*bi + g01i*br;
    re[j] = g10r*ar - g10i*ai + g11r*br - g11i*bi;
    im[j] = g10r*ai + g10i*ar + g11r*bi + g11i*br;
  }
}
__device__ inline void cnotg(float* re, float* im, int a, int b) {
  int cm = 1 << (4 - a), tm = 1 << (4 - b);
  for (int i = 0; i < 32; ++i) {
    if (!(i & cm) || (i & tm)) continue;
    int j = i | tm;
    float tr = re[i], ti = im[i];
    re[i] = re[j]; im[i] = im[j];
    re[j] = tr;    im[j] = ti;
  }
}
__device__ inline void czg(float* re, float* im, int a, int b) {
  int m = (1 << (4 - a)) | (1 << (4 - b));
  for (int i = 0; i < 32; ++i)
    if ((i & m) == m) { re[i] = -re[i]; im[i] = -im[i]; }
}
__device__ inline void rzg(float* re, float* im, int w, float t) {
  float c = cosf(0.5f * t), s = sinf(0.5f * t);
  apply1(re, im, w, c, -s, 0.f, 0.f, 0.f, 0.f, c, s);
}
__device__ inline void rxg(float* re, float* im, int w, float t) {
  float c = cosf(0.5f * t), s = sinf(0.5f * t);
  apply1(re, im, w, c, 0.f, 0.f, -s, 0.f, -s, c, 0.f);
}
__device__ inline void ryg(float* re, float* im, int w, float t) {
  float c = cosf(0.5f * t), s = sinf(0.5f * t);
  apply1(re, im, w, c, 0.f, -s, 0.f, s, 0.f, c, 0.f);
}
__device__ inline void cryg(float* re, float* im, int cw, int tw, float t) {
  float c = cosf(0.5f * t), s = sinf(0.5f * t);
  applyc1(re, im, cw, tw, c, 0.f, -s, 0.f, s, 0.f, c, 0.f);
}
__device__ inline void crxg(float* re, float* im, int cw, int tw, float t) {
  float c = cosf(0.5f * t), s = sinf(0.5f * t);
  applyc1(re, im, cw, tw, c, 0.f, 0.f, -s, 0.f, -s, c, 0.f);
}

__global__ void qcnn_precompute(const float* __restrict__ conv,
                                const float* __restrict__ pool,
                                const float* __restrict__ fcw,
                                float* __restrict__ G) {
  __shared__ float Ur[32 * LSTRIDE];
  __shared__ float Ui[32 * LSTRIDE];
  const float HPI = 1.5707963267948966f;
  int t = threadIdx.x;  // basis state / column index, 0..31
  float* re = &Ur[t * LSTRIDE];
  float* im = &Ui[t * LSTRIDE];
  for (int i = 0; i < 32; ++i) { re[i] = (i == t) ? 1.f : 0.f; im[i] = 0.f; }

  int wires[5] = {0, 1, 2, 3, 4};
  int nw = 5;
  for (int l = 0; l < 2; ++l) {
    // conv_block
    for (int i = 0; i < nw; ++i) {
      int a = wires[i], b = wires[(i + 1) % nw];
      cnotg(re, im, a, b);
      rzg(re, im, b, conv[l * 15 + i]);
      cnotg(re, im, a, b);
      rxg(re, im, a, HPI);  rxg(re, im, b, HPI);
      cnotg(re, im, a, b);
      rxg(re, im, a, -HPI); rxg(re, im, b, -HPI);
      czg(re, im, a, b);
      rzg(re, im, b, conv[l * 15 + i + 2]);
      czg(re, im, a, b);
    }
    // pool_block
    int a = wires[0], b = wires[1];
    ryg(re, im, a, pool[l * 2 + 0]);
    ryg(re, im, b, pool[l * 2 + 1]);
    cnotg(re, im, a, b);
    cryg(re, im, b, a, pool[l * 2 + 0]);  // control=b, target=a
    crxg(re, im, a, b, pool[l * 2 + 1]);  // control=a, target=b
    cnotg(re, im, b, a);
    for (int i = 0; i < nw - 1; ++i) wires[i] = wires[i + 1];
    nw -= 1;
  }
  __syncthreads();
  // measured wires after 2 layers: {2,3} -> flat bits 2 and 1
  // G_c[i][j] = sum_n (fcw[c][0]*s2(n)+fcw[c][1]*s3(n)) * Re(conj(U[n,i])U[n,j])
  for (int c = 0; c < 2; ++c) {
    float w0 = fcw[c * 2 + 0], w1 = fcw[c * 2 + 1];
    for (int j = 0; j < 32; ++j) {
      float acc = 0.f;
      for (int n = 0; n < 32; ++n) {
        float s2 = ((n >> 2) & 1) ? -1.f : 1.f;
        float s3 = ((n >> 1) & 1) ? -1.f : 1.f;
        float wc = w0 * s2 + w1 * s3;
        acc += wc * (Ur[t * LSTRIDE + n] * Ur[j * LSTRIDE + n] +
                     Ui[t * LSTRIDE + n] * Ui[j * LSTRIDE + n]);
      }
      G[c * 1024 + t * 32 + j] = acc;
    }
  }
}

// ================= Main kernel: batched quadratic forms via f32 WMMA =================
__device__ inline v8f wmma4(v2f a, v2f b, v8f c) {
  return __builtin_amdgcn_wmma_f32_16x16x4_f32(false, a, false, b, (short)0, c,
                                               false, false);
}

__global__ __launch_bounds__(256) void qcnn_main(const float* __restrict__ x,
                                                 const float* __restrict__ G,
                                                 const float* __restrict__ fcb,
                                                 float* __restrict__ out,
                                                 int nTiles) {
  __shared__ float Gs[2 * 32 * 32];
  __shared__ float vt[8][16 * LSTRIDE];

  int tid = threadIdx.x;
  for (int i = tid; i < 2048; i += 256) Gs[i] = G[i];
  __syncthreads();

  int wave = tid >> 5;
  int lane = tid & 31;
  int lm = lane & 15;   // lane within half
  int hi = lane >> 4;   // half-wave index
  float* v = vt[wave];

  // Preload B fragments of G: bf[class][n-tile][k-chunk]
  // B[k][n] layout: lane l, vgpr p holds G[k0 + (l<16?0:2) + p][n0 + l%16]
  v2f bf[2][2][8];
#pragma unroll
  for (int c = 0; c < 2; ++c)
#pragma unroll
    for (int t2 = 0; t2 < 2; ++t2)
#pragma unroll
      for (int k = 0; k < 8; ++k) {
        int row = k * 4 + hi * 2;
        int col = t2 * 16 + lm;
        bf[c][t2][k].x = Gs[c * 1024 + row * 32 + col];
        bf[c][t2][k].y = Gs[c * 1024 + (row + 1) * 32 + col];
      }
  float fb0 = fcb[0], fb1 = fcb[1];

  int globalWave = blockIdx.x * 8 + wave;
  int strideWaves = gridDim.x * 8;
  int iters = (nTiles + strideWaves - 1) / strideWaves;

  for (int it = 0; it < iters; ++it) {
    int tile = globalWave + it * strideWaves;
    bool active = tile < nTiles;   // wave-uniform
    int base = tile * 16;

    if (active) {
      // Build 16x32 amplitude tile: two lanes per batch row, 16 amps each.
      int b = base + lm;
      float cq[5], sq[5];
#pragma unroll
      for (int q = 0; q < 5; ++q) {
        float xv = 0.5f * x[b * 5 + q];
        sincosf(xv, &sq[q], &cq[q]);
      }
      float a0 = hi ? sq[0] : cq[0];  // qubit0 = bit4 of flat index
#pragma unroll
      for (int i1 = 0; i1 < 2; ++i1) {
        float a1 = a0 * (i1 ? sq[1] : cq[1]);
#pragma unroll
        for (int i2 = 0; i2 < 2; ++i2) {
          float a2 = a1 * (i2 ? sq[2] : cq[2]);
#pragma unroll
          for (int i3 = 0; i3 < 2; ++i3) {
            float a3 = a2 * (i3 ? sq[3] : cq[3]);
#pragma unroll
            for (int i4 = 0; i4 < 2; ++i4) {
              int idx = hi * 16 + i1 * 8 + i2 * 4 + i3 * 2 + i4;
              v[lm * LSTRIDE + idx] = a3 * (i4 ? sq[4] : cq[4]);
            }
          }
        }
      }
    }
    __syncthreads();

    if (active) {
      v8f z = {};
      v8f acc00 = z, acc01 = z, acc10 = z, acc11 = z;
#pragma unroll
      for (int k = 0; k < 8; ++k) {
        // A fragment: lane l holds V[base + l%16, k*4 + (l<16?0:2) + p]
        v2f a;
        int rb = k * 4 + hi * 2;
        a.x = v[lm * LSTRIDE + rb];
        a.y = v[lm * LSTRIDE + rb + 1];
        acc00 = wmma4(a, bf[0][0][k], acc00);
        acc01 = wmma4(a, bf[0][1][k], acc01);
        acc10 = wmma4(a, bf[1][0][k], acc10);
        acc11 = wmma4(a, bf[1][1][k], acc11);
      }
      // Finish: out_c[b] = sum_n Y_c[b,n] * V[b,n]; D element (r,lane) ->
      // (m = r + hi*8, n = lm) for tile0 and n = 16+lm for tile1.
      float p0[8], p1[8];
#pragma unroll
      for (int r = 0; r < 8; ++r) {
        int m = hi * 8 + r;
        float v0 = v[m * LSTRIDE + lm];
        float v1 = v[m * LSTRIDE + 16 + lm];
        p0[r] = acc00[r] * v0 + acc01[r] * v1;
        p1[r] = acc10[r] * v0 + acc11[r] * v1;
      }
#pragma unroll
      for (int off = 1; off < 16; off <<= 1) {
#pragma unroll
        for (int r = 0; r < 8; ++r) {
          p0[r] += __shfl_xor(p0[r], off, 32);
          p1[r] += __shfl_xor(p1[r], off, 32);
        }
      }
      if (lm == 0) {
#pragma unroll
        for (int r = 0; r < 8; ++r) {
          int bb = base + hi * 8 + r;
          out[bb * 2 + 0] = p0[r] + fb0;
          out[bb * 2 + 1] = p1[r] + fb1;
        }
      }
    }
    __syncthreads();  // vtile reused next iteration
  }
}

extern "C" void kernel_launch(void* const* d_in, const int* in_sizes, int n_in,
                              void* d_out, int out_size, void* d_ws, size_t ws_size,
                              hipStream_t stream) {
  const float* x    = (const float*)d_in[0];  // (B, 5)
  const float* conv = (const float*)d_in[1];  // (2, 15)
  const float* pool = (const float*)d_in[2];  // (2, 2)
  const float* fcw  = (const float*)d_in[3];  // (2, 2)
  const float* fcb  = (const float*)d_in[4];  // (2,)
  float* out = (float*)d_out;                 // (B, 2)
  float* G = (float*)d_ws;                    // 2*32*32 floats = 8 KB

  int B = in_sizes[0] / 5;
  qcnn_precompute<<<1, 32, 0, stream>>>(conv, pool, fcw, G);

  int nTiles = B / 16;  // B = 262144 -> 16384 tiles of 16 batch rows
  qcnn_main<<<256, 256, 0, stream>>>(x, G, fcb, out, nTiles);
}